// TiedTranspose_27616639713560
// MI455X (gfx1250) — compile-verified
//
#include <hip/hip_runtime.h>
#include <hip/hip_bf16.h>

typedef float v2f __attribute__((ext_vector_type(2)));
typedef float v8f __attribute__((ext_vector_type(8)));

#define Bm 16          // batch (M)
#define Lk 32768       // latents (K)
#define Dn 768         // d_model (N)
#define LCHUNK 512     // K-split chunk
#define NCH 64         // Lk / LCHUNK
#define NTILES 48      // Dn / 16
#define WPB 4          // waves per block (128 threads -> finer load balance)
#define NGROUPS 12     // NTILES / WPB
#define XS_STRIDE 516  // padded LDS row stride (floats) -> conflict-free b64 reads

// Kernel 1: per (lchunk, ntile-group) block, WMMA f32 16x16x4 partial GEMM.
__global__ __launch_bounds__(128) void sae_wmma_partial_kernel(
    const float* __restrict__ x, const float* __restrict__ W,
    float* __restrict__ partial)
{
  __shared__ float xs[Bm * XS_STRIDE];   // ~33 KB of the 320 KB WGP LDS

  const int lchunk = blockIdx.x & (NCH - 1);
  const int ngroup = blockIdx.x / NCH;
  const int l0 = lchunk * LCHUNK;
  const int t = threadIdx.x;

  // Stage x[0:16, l0:l0+512] into LDS: 2048 float4 tiles, 16 per thread.
  #pragma unroll
  for (int i = 0; i < 16; ++i) {
    const int linear = t + i * 128;
    const int m  = linear >> 7;          // / (LCHUNK/4)
    const int lq = linear & 127;
    const float4 v =
        *reinterpret_cast<const float4*>(x + m * Lk + l0 + (lq << 2));
    *reinterpret_cast<float4*>(&xs[m * XS_STRIDE + (lq << 2)]) = v;  // 16B aligned (516*4 % 16 == 0)
  }
  __syncthreads();

  const int wave  = t >> 5;
  const int lane  = t & 31;
  const int ntile = ngroup * WPB + wave;
  const int m     = lane & 15;           // A: M index on lanes; B: N index on lanes
  const int kk    = (lane >> 4) << 1;    // half-wave K offset: 0 or 2
  const int ncol  = ntile * 16 + m;
  const float* wcol = W + ncol;          // column base; row stride Dn

  v8f c = {};                            // 16x16 f32 accumulator (8 VGPRs)

  #pragma unroll 1
  for (int k = 0; k < LCHUNK; k += 16) {
    #pragma unroll
    for (int u = 0; u < 4; ++u) {
      const int kb = k + u * 4;
      // A 16x4: v0 = x[m][kb+kk], v1 = x[m][kb+kk+1]  (ds_load_2addr_b64 pairs)
      v2f a = *reinterpret_cast<const v2f*>(&xs[m * XS_STRIDE + kb + kk]);
      // B 4x16: v0 = W[l0+kb+kk][ncol], v1 = W[l0+kb+kk+1][ncol]
      v2f b;
      b.x = wcol[(l0 + kb + kk) * Dn];
      b.y = wcol[(l0 + kb + kk + 1) * Dn];
      c = __builtin_amdgcn_wmma_f32_16x16x4_f32(
          /*neg_a=*/false, a, /*neg_b=*/false, b,
          /*c_mod=*/(short)0, c, /*reuse_a=*/false, /*reuse_b=*/false);
    }
    // Prefetch W 64 rows ahead (global_prefetch_b8), clamped in-bounds.
    int pl = l0 + k + 64;
    if (pl > Lk - 1) pl = Lk - 1;
    __builtin_prefetch(wcol + pl * Dn, 0, 1);
  }

  // C/D layout: VGPR r -> M = r + 8*(lane>=16), N = lane&15 (+tile base).
  float* pt = partial + (size_t)lchunk * (Bm * Dn);
  const int mrow = (lane >> 4) << 3;     // 0 or 8
  #pragma unroll
  for (int r = 0; r < 8; ++r) {
    pt[(mrow + r) * Dn + ncol] = c[r];
  }
}

// Kernel 2: deterministic split-K reduction over the 64 partial tiles.
__global__ __launch_bounds__(256) void sae_reduce_kernel(
    const float* __restrict__ partial, float* __restrict__ out)
{
  const int idx = blockIdx.x * 256 + threadIdx.x;   // 0 .. Bm*Dn-1
  float s0 = 0.f, s1 = 0.f, s2 = 0.f, s3 = 0.f;
  #pragma unroll
  for (int cch = 0; cch < NCH; cch += 4) {
    s0 += partial[(size_t)(cch + 0) * (Bm * Dn) + idx];
    s1 += partial[(size_t)(cch + 1) * (Bm * Dn) + idx];
    s2 += partial[(size_t)(cch + 2) * (Bm * Dn) + idx];
    s3 += partial[(size_t)(cch + 3) * (Bm * Dn) + idx];
  }
  out[idx] = (s0 + s1) + (s2 + s3);
}

extern "C" void kernel_launch(void* const* d_in, const int* in_sizes, int n_in,
                              void* d_out, int out_size, void* d_ws, size_t ws_size,
                              hipStream_t stream) {
  const float* x = (const float*)d_in[0];       // [16, 32768] f32
  const float* W = (const float*)d_in[1];       // [32768, 768] f32
  float* out     = (float*)d_out;               // [16, 768] f32
  float* partial = (float*)d_ws;                // [64, 16, 768] f32 = 3 MB scratch

  sae_wmma_partial_kernel<<<NCH * NGROUPS, 128, 0, stream>>>(x, W, partial);
  sae_reduce_kernel<<<(Bm * Dn) / 256, 256, 0, stream>>>(partial, out);
}